// ResilienceRNN_15771119911593
// MI455X (gfx1250) — compile-verified
//
#include <hip/hip_runtime.h>
#include <math.h>

// ---- reference constants ----
#define OMEGA_U     0.0002073f
#define NOISE_LEVEL 0.05f
#define EPS_F       1e-8f

#define NWG 64      // persistent workgroups (must all be co-resident; MI455X has many more WGPs)
#define TPB 256     // 8 wave32 waves
#define DIM 4096    // problem dim fixed by the reference; compile-time for unrollable loops
#define KPH (DIM / 2)        // K elements per K-half (2048)
#define KITER (KPH / 8)      // iterations of the dual-accumulator loop (each does 2 x K=4)

typedef __attribute__((ext_vector_type(2))) float v2f;
typedef __attribute__((ext_vector_type(8))) float v8f;

// ---------------- init: zero accumulators, copy h0, reset barrier, precompute trig tables ----
__global__ void rnn_init_kernel(const float* __restrict__ h0, float* __restrict__ hbuf,
                                float* __restrict__ dotacc, float* __restrict__ nrmacc,
                                float* __restrict__ cost, float* __restrict__ sint,
                                unsigned* __restrict__ bar, int dim, int steps) {
  int i = blockIdx.x * blockDim.x + threadIdx.x;
  if (i < steps) {
    dotacc[i] = 0.0f;
    nrmacc[i] = 0.0f;
    float ph = OMEGA_U * (float)i;
    cost[i] = cosf(ph);            // full-accuracy libm trig, off the critical path
    sint[i] = sinf(ph);
  }
  if (i < dim) { hbuf[i] = h0[i]; }
  if (i == 0)  { bar[0] = 0u; bar[1] = 0u; }
}

// ---------------- ||h0|| (single workgroup reduction) ----------------
__global__ void rnn_h0norm_kernel(const float* __restrict__ h0, float* __restrict__ h0n, int dim) {
  __shared__ float red[TPB];
  float s = 0.0f;
  for (int i = threadIdx.x; i < dim; i += blockDim.x) { float v = h0[i]; s += v * v; }
  red[threadIdx.x] = s;
  __syncthreads();
  for (int off = TPB / 2; off > 0; off >>= 1) {
    if ((int)threadIdx.x < off) red[threadIdx.x] += red[threadIdx.x + off];
    __syncthreads();
  }
  if (threadIdx.x == 0) h0n[0] = fmaxf(sqrtf(red[0]), EPS_F);
}

// ---------------- persistent recurrence kernel ----------------
// Grid = NWG x TPB. WG wg owns columns [wg*64, wg*64+64) of h_new = h @ W.
// Wave w: tile = w&3 (16 cols), khalf = w>>2 (half of the K=4096 reduction).
// Inner op: V_WMMA_F32_16X16X4_F32 with A row0 = h[kb..kb+3], B = W[kb..kb+3][j0..j0+15].
// A-fragment masking is done by pointing non-head lanes at a zero slab in LDS, so the
// ds_load result feeds WMMA directly (no per-iteration cndmasks).
__global__ void __launch_bounds__(TPB) rnn_persistent_kernel(
    const float* __restrict__ x, const float* __restrict__ W,
    const float* __restrict__ h0, const float* __restrict__ noise,
    float* __restrict__ hbuf, float* __restrict__ dotacc, float* __restrict__ nrmacc,
    const float* __restrict__ h0n, const float* __restrict__ cost,
    const float* __restrict__ sint, unsigned* __restrict__ bar,
    float* __restrict__ sims, int steps) {

  __shared__ float hlds[DIM];         // current h, staged once per step
  __shared__ float hzero[DIM];        // permanent zeros: A-fragment source for non-head lanes
  __shared__ float partial[8 * 16];   // per-wave 16-column partial results

  const int tid   = threadIdx.x;
  const int wg    = blockIdx.x;
  const int lane  = tid & 31;
  const int wave  = tid >> 5;
  const int tile  = wave & 3;
  const int khalf = wave >> 2;

  const int colsPerWg = DIM / NWG;                 // 64
  const int j0        = wg * colsPerWg + tile * 16;
  const int kbase     = khalf * KPH;               // this wave's K-half start

  const int  half = lane >> 4;        // 0: lanes 0-15 (K0/K1), 1: lanes 16-31 (K2/K3)
  const int  nIdx = lane & 15;        // column within tile
  const bool head = (nIdx == 0);      // only lanes 0 and 16 carry A-matrix data (row M=0)
  const int  hoff = kbase + (half << 1);                  // per-lane h row base
  const float* wcol = W + (size_t)hoff * DIM + (size_t)(j0 + nIdx);
  const float* habase = head ? hlds : hzero;              // per-lane A-fragment source

  // zero the A-masking slab once (re-done every launch; deterministic)
  for (int i = tid; i < DIM; i += TPB) hzero[i] = 0.0f;

  for (int t = 0; t < steps; ++t) {
    const float* cur = hbuf + (size_t)(t & 1) * (size_t)DIM;
    float*       nxt = hbuf + (size_t)((t + 1) & 1) * (size_t)DIM;

    // stage h into LDS (read by all waves' A-fragments)
    for (int i = tid; i < DIM; i += TPB) hlds[i] = cur[i];
    __syncthreads();

    // ---- K-loop: f32 WMMA rank-4 updates; W streamed from the 192MB L2 ----
    v8f acc0 = {};                    // two chains to relax the WMMA RAW dependency
    v8f acc1 = {};
#pragma unroll 8
    for (int i = 0; i < KITER; ++i) {             // KITER=256, compile-time -> unrolls
      const int k0 = hoff + i * 8;                // chain 0: K rows k0.., chain 1: k0+4..
      v2f a0, a1, b0, b1;
      a0.x = habase[k0];                          // head lanes: h values; others: zeros
      a0.y = habase[k0 + 1];
      a1.x = habase[k0 + 4];
      a1.y = habase[k0 + 5];
      const float* p = wcol + (size_t)i * (8 * DIM);
      b0.x = p[0];                                // row k0   (B VGPR0: K0|K2)
      b0.y = p[DIM];                              // row k0+1 (B VGPR1: K1|K3)
      b1.x = p[4 * DIM];                          // row k0+4
      b1.y = p[5 * DIM];                          // row k0+5
      acc0 = __builtin_amdgcn_wmma_f32_16x16x4_f32(false, a0, false, b0,
                                                   (short)0, acc0, false, false);
      acc1 = __builtin_amdgcn_wmma_f32_16x16x4_f32(false, a1, false, b1,
                                                   (short)0, acc1, false, false);
    }

    if (lane < 16) partial[wave * 16 + lane] = acc0[0] + acc1[0];
    __syncthreads();

    // ---- epilogue: combine K-halves, add input, torsion rotate pairs, local reductions ----
    if (tid < 32) {
      const float c = cost[t], s = sint[t];       // precomputed, uniform (SMEM load)
      const int jle = 2 * tid, jlo = jle + 1;     // local even/odd columns (same tile)
      const int tl  = jle >> 4;
      const float ve = partial[tl * 16 + (jle & 15)] + partial[(tl + 4) * 16 + (jle & 15)];
      const float vo = partial[tl * 16 + (jlo & 15)] + partial[(tl + 4) * 16 + (jlo & 15)];
      const int jge = wg * colsPerWg + jle;
      const int jgo = jge + 1;
      const size_t nb = (size_t)t * (size_t)DIM;
      const float he = ve + 0.01f * (x[jge] + NOISE_LEVEL * noise[nb + jge]);
      const float ho = vo + 0.01f * (x[jgo] + NOISE_LEVEL * noise[nb + jgo]);
      const float ne = he * c - ho * s;           // even' = e*c - o*s
      const float no = he * s + ho * c;           // odd'  = e*s + o*c
      nxt[jge] = ne;
      nxt[jgo] = no;
      float d  = h0[jge] * ne + h0[jgo] * no;
      float nm = ne * ne + no * no;
      for (int off = 16; off > 0; off >>= 1) {    // wave32 reduction
        d  += __shfl_down(d,  (unsigned)off, 32);
        nm += __shfl_down(nm, (unsigned)off, 32);
      }
      if (tid == 0) {
        atomicAdd(&dotacc[t], d);                 // global_atomic_add_f32
        atomicAdd(&nrmacc[t], nm);
      }
    }

    // ---- grid-wide barrier (release before arrive, acquire after pass) ----
    __builtin_amdgcn_fence(__ATOMIC_RELEASE, "agent");
    __syncthreads();
    if (tid == 0) {
      const unsigned g = __hip_atomic_load(&bar[1], __ATOMIC_ACQUIRE, __HIP_MEMORY_SCOPE_AGENT);
      const unsigned arrived =
          __hip_atomic_fetch_add(&bar[0], 1u, __ATOMIC_ACQ_REL, __HIP_MEMORY_SCOPE_AGENT) + 1u;
      if (arrived == (unsigned)gridDim.x) {
        __hip_atomic_store(&bar[0], 0u, __ATOMIC_RELAXED, __HIP_MEMORY_SCOPE_AGENT);
        __hip_atomic_fetch_add(&bar[1], 1u, __ATOMIC_RELEASE, __HIP_MEMORY_SCOPE_AGENT);
      } else {
        while (__hip_atomic_load(&bar[1], __ATOMIC_ACQUIRE, __HIP_MEMORY_SCOPE_AGENT) == g) {
          __builtin_amdgcn_s_sleep(1);
        }
      }
    }
    __syncthreads();
    __builtin_amdgcn_fence(__ATOMIC_ACQUIRE, "agent");

    // accumulators for step t are final after the barrier
    if (wg == 0 && tid == 0) {
      const float hn = fmaxf(sqrtf(nrmacc[t]), EPS_F);
      sims[t] = dotacc[t] / (h0n[0] * hn);
    }
  }
}

// ---------------- host launcher ----------------
extern "C" void kernel_launch(void* const* d_in, const int* in_sizes, int n_in,
                              void* d_out, int out_size, void* d_ws, size_t ws_size,
                              hipStream_t stream) {
  const float* x     = (const float*)d_in[0];   // [1, dim]
  const float* W     = (const float*)d_in[1];   // [dim, dim] row-major
  const float* h0    = (const float*)d_in[2];   // [1, dim]
  const float* noise = (const float*)d_in[3];   // [steps, 1, dim]
  const int dim   = in_sizes[0];                // 4096 (kernel assumes DIM)
  const int steps = in_sizes[3] / dim;          // 3000
  float* sims = (float*)d_out;

  // workspace layout
  float*    hbuf   = (float*)d_ws;              // 2*DIM (ping-pong h)
  float*    dotacc = hbuf + 2 * DIM;            // steps
  float*    nrmacc = dotacc + steps;            // steps
  float*    cost   = nrmacc + steps;            // steps
  float*    sint   = cost + steps;              // steps
  float*    h0n    = sint + steps;              // 1
  unsigned* bar    = (unsigned*)(h0n + 1);      // [count, generation]

  const int initN = (steps > dim) ? steps : dim;
  rnn_init_kernel<<<(initN + TPB - 1) / TPB, TPB, 0, stream>>>(h0, hbuf, dotacc, nrmacc,
                                                               cost, sint, bar, dim, steps);
  rnn_h0norm_kernel<<<1, TPB, 0, stream>>>(h0, h0n, dim);
  rnn_persistent_kernel<<<NWG, TPB, 0, stream>>>(x, W, h0, noise, hbuf, dotacc, nrmacc,
                                                 h0n, cost, sint, bar, sims, steps);
}